// PatchAttentionWithPairBias_88862873354621
// MI455X (gfx1250) — compile-verified
//
#include <hip/hip_runtime.h>

typedef __attribute__((ext_vector_type(2))) float v2f;
typedef __attribute__((ext_vector_type(8))) float v8f;

#define ATT_DIM   256
#define NUM_HEADS 8
#define HEAD_DIM  32
#define PAIR_DIM  64
#define PDIM      512
#define BSZ       8
#define LN_EPS    1e-3f

__device__ __forceinline__ float wave_sum(float v) {
#pragma unroll
  for (int o = 16; o > 0; o >>= 1) v += __shfl_xor(v, o, 32);
  return v;
}

// ---------------------------------------------------------------------------
// Kernel 1: Fn = LayerNorm(F) * g + b     rows = B*P = 4096, C = 256
// ---------------------------------------------------------------------------
__global__ __launch_bounds__(256) void ln_feat_kernel(
    const float* __restrict__ F, const float* __restrict__ g,
    const float* __restrict__ bta, float* __restrict__ Fn) {
  int row  = blockIdx.x * 8 + (threadIdx.x >> 5);
  int lane = threadIdx.x & 31;
  const float4* src = (const float4*)(F + (size_t)row * ATT_DIM);
  float4 a = src[lane];
  float4 c = src[lane + 32];
  float mu = wave_sum(a.x + a.y + a.z + a.w + c.x + c.y + c.z + c.w) * (1.f / 256.f);
  float d0 = a.x - mu, d1 = a.y - mu, d2 = a.z - mu, d3 = a.w - mu;
  float e0 = c.x - mu, e1 = c.y - mu, e2 = c.z - mu, e3 = c.w - mu;
  float var = wave_sum(d0*d0 + d1*d1 + d2*d2 + d3*d3 +
                       e0*e0 + e1*e1 + e2*e2 + e3*e3) * (1.f / 256.f);
  float inv = rsqrtf(var + LN_EPS);
  int c0 = lane * 4, c1 = 128 + lane * 4;
  float4 o1, o2;
  o1.x = d0 * inv * g[c0 + 0] + bta[c0 + 0];
  o1.y = d1 * inv * g[c0 + 1] + bta[c0 + 1];
  o1.z = d2 * inv * g[c0 + 2] + bta[c0 + 2];
  o1.w = d3 * inv * g[c0 + 3] + bta[c0 + 3];
  o2.x = e0 * inv * g[c1 + 0] + bta[c1 + 0];
  o2.y = e1 * inv * g[c1 + 1] + bta[c1 + 1];
  o2.z = e2 * inv * g[c1 + 2] + bta[c1 + 2];
  o2.w = e3 * inv * g[c1 + 3] + bta[c1 + 3];
  float4* dst = (float4*)(Fn + (size_t)row * ATT_DIM);
  dst[lane]      = o1;
  dst[lane + 32] = o2;
}

// ---------------------------------------------------------------------------
// Kernel 2: pair bias via WMMA.
// Bias[b,h,p,q] = sum_c relu(LN(D[b,p,q,:]))[c] * Wp[c,h]
// A 16-pair tile of relu(LN(D)) is built directly in f32 A-fragment layout
// (lane l & l+16 own pair-row l; k-step kk covers channels 4kk..4kk+3), and
// Wp (64x8, zero-padded to N=16) is 16 hoisted loop-invariant B-fragments.
// 16 WMMAs replace the 512-FMA + 40-shuffle per-tile reduction.
// D is read once (536 MB) -> non-temporal loads keep L2 free for Bias/QKVG.
// ---------------------------------------------------------------------------
__global__ __launch_bounds__(256) void pair_bias_kernel(
    const float* __restrict__ D, const float* __restrict__ g,
    const float* __restrict__ bta, const float* __restrict__ Wp,
    float* __restrict__ Bias) {
  const int TPW = 8;                       // tiles per wave
  int wid  = blockIdx.x * 8 + (threadIdx.x >> 5);
  int lane = threadIdx.x & 31;
  int half = lane >> 4;                    // 0: K=4kk,4kk+1   1: K=4kk+2,4kk+3
  int l    = lane & 15;
  int n    = l;                            // output column (head), valid < 8

  // hoisted per-lane constants: gamma/beta pairs and Wp B-fragments
  v2f g2[16], b2[16], wf[16];
#pragma unroll
  for (int kk = 0; kk < 16; ++kk) {
    int c = kk * 4 + half * 2;
    g2[kk].x = g[c];     g2[kk].y = g[c + 1];
    b2[kk].x = bta[c];   b2[kk].y = bta[c + 1];
    wf[kk].x = (n < NUM_HEADS) ? Wp[c * NUM_HEADS + n]       : 0.f;
    wf[kk].y = (n < NUM_HEADS) ? Wp[(c + 1) * NUM_HEADS + n] : 0.f;
  }

  for (int t = 0; t < TPW; ++t) {
    long long tile  = (long long)wid * TPW + t;     // 0 .. 131071
    long long pair0 = tile << 4;                    // first (b,p,q) pair
    const v2f* row = (const v2f*)(D + (pair0 + l) * PAIR_DIM);
    v2f x[16];
#pragma unroll
    for (int kk = 0; kk < 16; ++kk)
      x[kk] = __builtin_nontemporal_load(&row[kk * 2 + half]);

    // LayerNorm over 64 channels of pair-row l (lanes l, l+16 hold halves)
    float s = 0.f;
#pragma unroll
    for (int kk = 0; kk < 16; ++kk) s += x[kk].x + x[kk].y;
    s += __shfl_xor(s, 16, 32);
    float mu = s * (1.f / 64.f);
    float vv = 0.f;
#pragma unroll
    for (int kk = 0; kk < 16; ++kk) {
      x[kk].x -= mu; x[kk].y -= mu;
      vv += x[kk].x * x[kk].x + x[kk].y * x[kk].y;
    }
    vv += __shfl_xor(vv, 16, 32);
    float inv = rsqrtf(vv * (1.f / 64.f) + LN_EPS);

    // relu(LN) -> A-fragment, multiply by Wp B-fragments
    v8f acc = {};
#pragma unroll
    for (int kk = 0; kk < 16; ++kk) {
      v2f y;
      y.x = fmaxf(x[kk].x * inv * g2[kk].x + b2[kk].x, 0.f);
      y.y = fmaxf(x[kk].y * inv * g2[kk].y + b2[kk].y, 0.f);
      acc = __builtin_amdgcn_wmma_f32_16x16x4_f32(false, y, false, wf[kk],
                                                  (short)0, acc, false, false);
    }

    // D layout: acc[r] -> pair q0 + r + half*8, head n. 8 contiguous q's.
    if (n < NUM_HEADS) {
      int q0 = (int)(pair0 & 511);
      int p  = (int)((pair0 >> 9) & 511);
      int b  = (int)(pair0 >> 18);
      float* dst = Bias + (((size_t)(b * NUM_HEADS + n) * PDIM + p) << 9)
                        + q0 + half * 8;
      ((float4*)dst)[0] = make_float4(acc[0], acc[1], acc[2], acc[3]);
      ((float4*)dst)[1] = make_float4(acc[4], acc[5], acc[6], acc[7]);
    }
  }
}

// ---------------------------------------------------------------------------
// WMMA f32 GEMM: C = A[MxK] @ W[KxN] (+ epilogue), one 16x16 tile per wave.
// mode 0: store head-split [b,h,p,d]        (Q, K, V)
// mode 1: sigmoid(x + bias[n]), head-split  (G)
// mode 2: x + bias[n], row-major            (final projection)
// ---------------------------------------------------------------------------
__global__ __launch_bounds__(128) void gemm_kernel(
    const float* __restrict__ A, const float* __restrict__ W,
    const float* __restrict__ bias, float* __restrict__ C,
    int M, int N, int K, int mode) {
  int wid    = blockIdx.x * (blockDim.x >> 5) + (threadIdx.x >> 5);
  int tilesN = N >> 4;
  int tm = (wid / tilesN) << 4;
  int tn = (wid % tilesN) << 4;
  int lane = threadIdx.x & 31;
  int half = lane >> 4;
  int l    = lane & 15;
  const float* Arow = A + (size_t)(tm + l) * K;
  v8f acc = {};
#pragma unroll 4
  for (int k = 0; k < K; k += 4) {
    int ka = k + 2 * half;
    v2f af, bf;
    af.x = Arow[ka];
    af.y = Arow[ka + 1];
    bf.x = W[(size_t)ka * N + tn + l];
    bf.y = W[(size_t)(ka + 1) * N + tn + l];
    acc = __builtin_amdgcn_wmma_f32_16x16x4_f32(false, af, false, bf,
                                                (short)0, acc, false, false);
  }
#pragma unroll
  for (int r = 0; r < 8; ++r) {
    int m = tm + r + half * 8;
    int n = tn + l;
    float v = acc[r];
    if (mode == 2) {
      C[(size_t)m * N + n] = v + bias[n];
    } else {
      if (mode == 1) v = 1.f / (1.f + __expf(-(v + bias[n])));
      int bb = m >> 9, p = m & 511, h = n >> 5, d = n & 31;
      C[(((size_t)(bb * NUM_HEADS + h) * PDIM) + p) * HEAD_DIM + d] = v;
    }
  }
}

// ---------------------------------------------------------------------------
// Kernel 4: attention core per (b,h, 16-query block).
// ---------------------------------------------------------------------------
__global__ __launch_bounds__(32) void attention_kernel(
    const float* __restrict__ Q, const float* __restrict__ Kh,
    const float* __restrict__ V, const float* __restrict__ G,
    const float* __restrict__ Bias, const unsigned char* __restrict__ mask,
    float* __restrict__ attn) {
  __shared__ float s[16 * 520];   // 33,280 B; stride 520 avoids bank conflicts
  int task = blockIdx.x;          // 0..2047
  int pblk = task & 31;
  int bh   = task >> 5;
  int b    = bh >> 3;
  int h    = bh & 7;
  int lane = threadIdx.x & 31;
  int half = lane >> 4;
  int l    = lane & 15;
  int m0   = pblk * 16;
  const float* Qb = Q    + (size_t)bh * PDIM * HEAD_DIM;
  const float* Kb = Kh   + (size_t)bh * PDIM * HEAD_DIM;
  const float* Vb = V    + (size_t)bh * PDIM * HEAD_DIM;
  const float* Gb = G    + (size_t)bh * PDIM * HEAD_DIM;
  const float* Bb = Bias + (size_t)bh * PDIM * PDIM;
  const unsigned char* mb = mask + b * PDIM;
  const float scale = 0.17677669529663687f;  // 1/sqrt(32)

  // ---- Phase 1: score strip [16 x 512] into LDS
  for (int qt = 0; qt < PDIM / 16; ++qt) {
    int qbase = qt * 16;
    v8f acc = {};
#pragma unroll
    for (int k = 0; k < HEAD_DIM; k += 4) {
      int ka = k + 2 * half;
      v2f af, bf;
      af.x = Qb[(m0 + l) * HEAD_DIM + ka];
      af.y = Qb[(m0 + l) * HEAD_DIM + ka + 1];
      bf.x = Kb[(qbase + l) * HEAD_DIM + ka];
      bf.y = Kb[(qbase + l) * HEAD_DIM + ka + 1];
      acc = __builtin_amdgcn_wmma_f32_16x16x4_f32(false, af, false, bf,
                                                  (short)0, acc, false, false);
    }
#pragma unroll
    for (int r = 0; r < 8; ++r) {
      int rr = r + half * 8;
      int q  = qbase + l;
      float v = acc[r] * scale + Bb[(m0 + rr) * PDIM + q];
      v = mb[q] ? v : -1e9f;
      s[rr * 520 + q] = v;
    }
  }
  __syncthreads();

  // ---- Phase 2: softmax; lanes (l, l+16) split row l into 256-col halves
  {
    float* row = s + l * 520 + half * 256;
    float mx = -1e30f;
    for (int i = 0; i < 256; ++i) mx = fmaxf(mx, row[i]);
    mx = fmaxf(mx, __shfl_xor(mx, 16, 32));
    float sum = 0.f;
    for (int i = 0; i < 256; ++i) {
      float e = __expf(row[i] - mx);
      row[i] = e;
      sum += e;
    }
    sum += __shfl_xor(sum, 16, 32);
    float invs = 1.f / sum;
    for (int i = 0; i < 256; ++i) row[i] *= invs;
  }
  __syncthreads();

  // ---- Phase 3: out = (w @ V) * sigmoidG, store [b,p, h*32+d]
  for (int dt = 0; dt < 2; ++dt) {
    v8f acc = {};
    for (int k = 0; k < PDIM; k += 4) {
      int ka = k + 2 * half;
      v2f af, bf;
      af.x = s[l * 520 + ka];
      af.y = s[l * 520 + ka + 1];
      bf.x = Vb[ka * HEAD_DIM + dt * 16 + l];
      bf.y = Vb[(ka + 1) * HEAD_DIM + dt * 16 + l];
      acc = __builtin_amdgcn_wmma_f32_16x16x4_f32(false, af, false, bf,
                                                  (short)0, acc, false, false);
    }
#pragma unroll
    for (int r = 0; r < 8; ++r) {
      int rr = r + half * 8;
      int p  = m0 + rr;
      int d  = dt * 16 + l;
      float v = Gb[p * HEAD_DIM + d] * acc[r];
      attn[((size_t)(b * PDIM + p)) * ATT_DIM + h * HEAD_DIM + d] = v;
    }
  }
}

// ---------------------------------------------------------------------------
extern "C" void kernel_launch(void* const* d_in, const int* in_sizes, int n_in,
                              void* d_out, int out_size, void* d_ws, size_t ws_size,
                              hipStream_t stream) {
  const float* F       = (const float*)d_in[0];
  const float* D       = (const float*)d_in[1];
  const unsigned char* mask = (const unsigned char*)d_in[2];
  const float* Wq      = (const float*)d_in[3];
  const float* Wk      = (const float*)d_in[4];
  const float* Wv      = (const float*)d_in[5];
  const float* Wg      = (const float*)d_in[6];
  const float* bg      = (const float*)d_in[7];
  const float* Wp      = (const float*)d_in[8];
  const float* Wo      = (const float*)d_in[9];
  const float* bo      = (const float*)d_in[10];
  const float* g_pair  = (const float*)d_in[11];
  const float* b_pair  = (const float*)d_in[12];
  const float* g_feat  = (const float*)d_in[13];
  const float* b_feat  = (const float*)d_in[14];
  float* out = (float*)d_out;

  // workspace layout (floats): Fn,Q,K,V,G,attn each 4 MB; Bias 64 MB
  float* ws   = (float*)d_ws;
  const size_t SEG = (size_t)1 << 20;   // 4096*256
  float* Fn   = ws;
  float* Qb   = ws + 1 * SEG;
  float* Kb   = ws + 2 * SEG;
  float* Vb   = ws + 3 * SEG;
  float* Gb   = ws + 4 * SEG;
  float* attn = ws + 5 * SEG;
  float* Bias = ws + 6 * SEG;           // 16 * SEG floats

  ln_feat_kernel<<<512, 256, 0, stream>>>(F, g_feat, b_feat, Fn);
  // 131072 tiles / 8 per wave / 8 waves per block = 2048 blocks
  pair_bias_kernel<<<2048, 256, 0, stream>>>(D, g_pair, b_pair, Wp, Bias);
  gemm_kernel<<<1024, 128, 0, stream>>>(Fn, Wq, nullptr, Qb, 4096, 256, 256, 0);
  gemm_kernel<<<1024, 128, 0, stream>>>(Fn, Wk, nullptr, Kb, 4096, 256, 256, 0);
  gemm_kernel<<<1024, 128, 0, stream>>>(Fn, Wv, nullptr, Vb, 4096, 256, 256, 0);
  gemm_kernel<<<1024, 128, 0, stream>>>(Fn, Wg, bg,      Gb, 4096, 256, 256, 1);
  attention_kernel<<<2048, 32, 0, stream>>>(Qb, Kb, Vb, Gb, Bias, mask, attn);
  gemm_kernel<<<1024, 128, 0, stream>>>(attn, Wo, bo, out, 4096, 256, 256, 2);
}